// Upsample_39616778338566
// MI455X (gfx1250) — compile-verified
//
#include <hip/hip_runtime.h>

// ---------------------------------------------------------------------------
// Sparse point conv on MI455X (gfx1250, wave32).
//
// Strategy: bin edges by kernel cell k (counting sort, 27 bins), then for each
// group run a fused gather->WMMA-GEMM->atomic-scatter:
//     out[i_high[e]] += x_low[j_low[e]] @ W_k^T
// Edge-wise GEMM = 32.8 GFLOP (less than the reference's dense 44.2 GFLOP
// transform) and, crucially, never materializes the 691 MB dense y tensor.
// x (25.6 MB), w (1.8 MB) and out (102 MB) all fit in the 192 MB L2, so HBM
// traffic collapses to ~150 MB.
//
// Matrix op: V_WMMA_F32_16X16X4_F32 (the only f32 WMMA) — reference is f32.
// Each block (4 waves) owns one 16-edge tile at a time; wave w owns output
// columns [32w, 32w+32). W_k columns for a wave are preloaded once per block
// into 64 register-resident B fragments and reused across all its tiles.
// ---------------------------------------------------------------------------

#define C_IN  128
#define C_OUT 128

typedef float v2f __attribute__((ext_vector_type(2)));
typedef float v8f __attribute__((ext_vector_type(8)));

__global__ void zero_counts_kernel(int* __restrict__ cnt, int K) {
    int t = threadIdx.x;
    if (t < K) cnt[t] = 0;
}

__global__ void hist_kernel(const int* __restrict__ kcell, int E,
                            int* __restrict__ cnt) {
    int e = blockIdx.x * 256 + threadIdx.x;
    if (e < E) atomicAdd(&cnt[kcell[e]], 1);
}

__global__ void prefix_kernel(const int* __restrict__ cnt, int K,
                              int* __restrict__ off, int* __restrict__ cur) {
    if (threadIdx.x == 0) {
        int acc = 0;
        for (int i = 0; i < K; ++i) { off[i] = acc; cur[i] = acc; acc += cnt[i]; }
        off[K] = acc;
    }
}

__global__ void sortscatter_kernel(const int* __restrict__ kcell, int E,
                                   int* __restrict__ cur, int* __restrict__ sorted) {
    int e = blockIdx.x * 256 + threadIdx.x;
    if (e < E) {
        int p = atomicAdd(&cur[kcell[e]], 1);
        sorted[p] = e;
    }
}

__global__ void bias_init_kernel(float* __restrict__ out,
                                 const float* __restrict__ bias, int total) {
    int t = blockIdx.x * 256 + threadIdx.x;
    if (t < total) out[t] = bias[t & (C_OUT - 1)];
}

// grid = (SPLITS, K); block = 128 (4 wave32)
__global__ __launch_bounds__(128)
void fused_pointconv_kernel(const float* __restrict__ x,
                            const float* __restrict__ w,
                            const int* __restrict__ i_high,
                            const int* __restrict__ j_low,
                            const int* __restrict__ off,
                            const int* __restrict__ sorted,
                            float* __restrict__ out) {
    const int kcell     = blockIdx.y;
    const int seg_begin = off[kcell];
    const int seg_end   = off[kcell + 1];
    if (seg_begin >= seg_end) return;
    const int ntiles = (seg_end - seg_begin + 15) >> 4;

    const int lane = threadIdx.x & 31;
    const int wave = threadIdx.x >> 5;
    const int lh   = lane & 15;            // row (A) / col-in-tile (B,C)
    const int koff = (lane >> 4) << 1;     // K sub-offset: lanes 0-15 -> k,k+1 ; 16-31 -> k+2,k+3

    // ---- Preload this wave's 32 columns of W_k as WMMA B fragments (regs) ---
    // B[c][n] = w[kcell][n][c]; frag s covers contraction steps c = 4s..4s+3.
    const float* wk = w + (size_t)kcell * (C_OUT * C_IN);
    const int col0 = wave * 32 + lh;       // first col tile
    v2f bfrag[64];
#pragma unroll
    for (int s = 0; s < 32; ++s) {
        const int cbase = s * 4 + koff;
        bfrag[2 * s + 0] = *(const v2f*)(wk + (size_t)(col0)      * C_IN + cbase);
        bfrag[2 * s + 1] = *(const v2f*)(wk + (size_t)(col0 + 16) * C_IN + cbase);
    }

    __shared__ int sm_j[16];
    __shared__ int sm_i[16];

    for (int tile = blockIdx.x; tile < ntiles; tile += gridDim.x) {
        const int base = seg_begin + tile * 16;
        if (threadIdx.x < 16) {
            const int e = base + threadIdx.x;
            if (e < seg_end) {
                const int se = sorted[e];
                sm_j[threadIdx.x] = j_low[se];
                sm_i[threadIdx.x] = i_high[se];
            } else {
                sm_j[threadIdx.x] = 0;
                sm_i[threadIdx.x] = -1;     // masked row
            }
        }
        __syncthreads();

        // A fragment row for this lane: gathered x row (L2-resident)
        const float* xrow = x + (size_t)sm_j[lh] * C_IN;

        // 4 independent accumulation chains (2 col tiles x even/odd parity)
        v8f acc00 = {}, acc01 = {}, acc10 = {}, acc11 = {};
#pragma unroll
        for (int s = 0; s < 32; ++s) {
            const v2f a = *(const v2f*)(xrow + s * 4 + koff);
            if (s & 1) {
                acc01 = __builtin_amdgcn_wmma_f32_16x16x4_f32(
                    false, a, false, bfrag[2 * s + 0], (short)0, acc01, false, false);
                acc11 = __builtin_amdgcn_wmma_f32_16x16x4_f32(
                    false, a, false, bfrag[2 * s + 1], (short)0, acc11, false, false);
            } else {
                acc00 = __builtin_amdgcn_wmma_f32_16x16x4_f32(
                    false, a, false, bfrag[2 * s + 0], (short)0, acc00, false, false);
                acc10 = __builtin_amdgcn_wmma_f32_16x16x4_f32(
                    false, a, false, bfrag[2 * s + 1], (short)0, acc10, false, false);
            }
        }
        const v8f acc0 = acc00 + acc01;    // tile 0 (cols col0)
        const v8f acc1 = acc10 + acc11;    // tile 1 (cols col0+16)

        // C layout: lanes 0-15 hold rows 0-7 in v0..v7, lanes 16-31 rows 8-15.
        const int rbase = (lane >> 4) << 3;
#pragma unroll
        for (int r = 0; r < 8; ++r) {
            const int row = rbase + r;
            const int ii  = sm_i[row];
            if (ii >= 0) {
                float* o = out + (size_t)ii * C_OUT + col0;
                __hip_atomic_fetch_add(o,      acc0[r], __ATOMIC_RELAXED,
                                       __HIP_MEMORY_SCOPE_AGENT);
                __hip_atomic_fetch_add(o + 16, acc1[r], __ATOMIC_RELAXED,
                                       __HIP_MEMORY_SCOPE_AGENT);
            }
        }
        __syncthreads();   // protect sm_j/sm_i before next tile
    }
}

extern "C" void kernel_launch(void* const* d_in, const int* in_sizes, int n_in,
                              void* d_out, int out_size, void* d_ws, size_t ws_size,
                              hipStream_t stream) {
    const float* x     = (const float*)d_in[0];   // [N_low, 128]
    const float* w     = (const float*)d_in[1];   // [K, 128, 128]
    const float* bias  = (const float*)d_in[2];   // [128]
    const int* i_high  = (const int*)d_in[3];     // [E]
    const int* j_low   = (const int*)d_in[4];     // [E]
    const int* kcell   = (const int*)d_in[5];     // [E]
    float* out = (float*)d_out;                   // [N_high, 128]

    const int E = in_sizes[3];
    const int K = in_sizes[1] / (C_IN * C_OUT);   // 27

    // Workspace layout (ints): cnt[32] | off[32] | cur[32] | sorted[E]
    int* ws_i   = (int*)d_ws;
    int* cnt    = ws_i;
    int* off    = ws_i + 32;
    int* cur    = ws_i + 64;
    int* sorted = ws_i + 96;

    const int eb = (E + 255) / 256;
    zero_counts_kernel<<<1, 64, 0, stream>>>(cnt, K);
    hist_kernel<<<eb, 256, 0, stream>>>(kcell, E, cnt);
    prefix_kernel<<<1, 32, 0, stream>>>(cnt, K, off, cur);
    sortscatter_kernel<<<eb, 256, 0, stream>>>(kcell, E, cur, sorted);
    bias_init_kernel<<<(out_size + 255) / 256, 256, 0, stream>>>(out, bias, out_size);

    dim3 grid(80, K);  // 80 splits per kernel cell -> 2160 workgroups
    fused_pointconv_kernel<<<grid, 128, 0, stream>>>(x, w, i_high, j_low,
                                                     off, sorted, out);
}